// GumbelSinkhornPermutation_3736621548274
// MI455X (gfx1250) — compile-verified
//
#include <hip/hip_runtime.h>
#include <stdint.h>

// ---------------------------------------------------------------------------
// Gumbel-Sinkhorn (B=16, N=256, TAU=0.1, 100 iters) for MI455X (gfx1250).
//
// Strategy: one workgroup (1024 threads = 32 wave32) per batch matrix.
// The whole 256KB f32 matrix lives in LDS (320KB/WGP on CDNA5) for all 100
// iterations -> HBM traffic is the 4MB load + 4MB store only.
//  * TDM tensor_load_to_lds with hardware row padding (pitch 257 floats)
//    => bank-conflict-free row AND column access.
//  * logsumexp sums via V_WMMA_F32_16X16X4_F32 against a ones operand:
//    D = E x 1 gives 16 row sums, D = 1 x E gives 16 col sums. With a ones
//    operand the result is invariant to the A/B K-lane mapping; only the
//    documented C/D layout is relied upon for extraction.
//  * exp/log on the trans pipe co-execute with WMMA accumulation.
// ---------------------------------------------------------------------------

typedef __attribute__((ext_vector_type(2))) float v2f;
typedef __attribute__((ext_vector_type(8))) float v8f;
typedef __attribute__((ext_vector_type(4))) unsigned int v4u;
typedef __attribute__((ext_vector_type(8))) int v8i;
typedef __attribute__((ext_vector_type(4))) int v4i;

#define NDIM 256
#define PITCH 257         // 257 % 64 == 1 -> column walks hit consecutive banks
#define TAU_INV 10.0f
#define SINK_ITERS 100

__device__ __forceinline__ v8f wmma_f32_16x16x4(v2f a, v2f b, v8f c) {
  // 8 args: (neg_a, A, neg_b, B, c_mod, C, reuse_a, reuse_b)
  return __builtin_amdgcn_wmma_f32_16x16x4_f32(false, a, false, b, (short)0, c,
                                               false, false);
}

__global__ __launch_bounds__(1024, 1)
void GumbelSinkhorn_gfx1250_kernel(const float* __restrict__ alpha,
                                   float* __restrict__ out) {
  __shared__ float tile[NDIM * PITCH];   // 264,192 B  (< 320 KB WGP LDS)
  __shared__ float lse_buf[NDIM];

  const int t    = threadIdx.x;
  const int b    = blockIdx.x;
  const int wv   = t >> 5;      // wave id (wave32)
  const int lane = t & 31;

  // ---- TDM: DMA the dense 256x256 f32 matrix into LDS, padding 1 dword
  // ---- after every 256 dwords so rows land at pitch 257.
  if (wv == 0) {
    uint64_t ga = (uint64_t)(uintptr_t)(alpha + (size_t)b * (NDIM * NDIM));
    uint32_t lds_base = (uint32_t)(uintptr_t)(void*)&tile[0];

    v4u g0;
    g0[0] = 1u;                                   // count=1, user descriptor
    g0[1] = lds_base;                             // lds_addr (bytes)
    g0[2] = (uint32_t)ga;                         // global_addr[31:0]
    g0[3] = ((uint32_t)(ga >> 32) & 0x01FFFFFFu)  // global_addr[56:32]
            | 0x80000000u;                        // type=2 ("image")

    v8i g1;
    g1[0] = (int)((2u << 16)      // data_size = 4B
                  | (1u << 20)    // pad_enable
                  | (7u << 22));  // pad_interval = 256 dwords (pad_amount=1)
    g1[1] = (int)(256u << 16);    // tensor_dim0 = 256
    g1[2] = (int)(256u << 16);    // tensor_dim1 = 256
    g1[3] = (int)(256u << 16);    // tile_dim0   = 256
    g1[4] = (int)256u;            // tile_dim1   = 256 (tile_dim2 = 0)
    g1[5] = (int)256u;            // tensor_dim0_stride = 256 elements
    g1[6] = 0;                    // tensor_dim1_stride[15:0] (65536 -> 0)
    g1[7] = 1;                    // tensor_dim1_stride[47:16] (65536 >> 16)

    v4i g2; g2[0] = 1; g2[1] = 1; g2[2] = 0; g2[3] = 0;  // dim2/3 benign
    v4i g3; g3[0] = 0; g3[1] = 0; g3[2] = 0; g3[3] = 0;

#if __has_include(<hip/amd_detail/amd_gfx1250_TDM.h>)
    v8i g4 = {};
    __builtin_amdgcn_tensor_load_to_lds(g0, g1, g2, g3, g4, 0);
#else
    __builtin_amdgcn_tensor_load_to_lds(g0, g1, g2, g3, 0);
#endif
    __builtin_amdgcn_s_wait_tensorcnt(0);
  }
  __syncthreads();

  // ---- scale by 1/TAU in place (lane-contiguous columns: conflict-free)
  {
    const int c  = t & (NDIM - 1);
    const int r0 = (t >> 8) * 64;
#pragma unroll 4
    for (int r = r0; r < r0 + 64; ++r)
      tile[r * PITCH + c] *= TAU_INV;
  }
  __syncthreads();

  const v2f vones = {1.0f, 1.0f};

  for (int it = 0; it < SINK_ITERS; ++it) {
    // ================= row normalization (logsumexp over columns) ========
    // Waves 0..15: wave w owns rows [16w, 16w+16). A-operand = exp(row data),
    // B = ones -> D[m][n] = rowsum_m. Two accumulator chains for ILP.
    if (wv < 16) {
      const int r0 = wv * 16;
      const int m  = lane & 15;
      const int kh = lane >> 4;  // which K half this lane feeds
      const float* rowp = &tile[(r0 + m) * PITCH + 2 * kh];
      v8f acc0 = {}; v8f acc1 = {};
#pragma unroll 4
      for (int k = 0; k < NDIM; k += 8) {
        v2f a0, a1;
        a0.x = __expf(rowp[k + 0]);
        a0.y = __expf(rowp[k + 1]);
        a1.x = __expf(rowp[k + 4]);
        a1.y = __expf(rowp[k + 5]);
        acc0 = wmma_f32_16x16x4(a0, vones, acc0);
        acc1 = wmma_f32_16x16x4(a1, vones, acc1);
      }
      v8f s = acc0 + acc1;
      // D layout: vgpr g, lanes 0-15 -> M=g ; lanes 16-31 -> M=8+g.
      if ((lane & 15) == 0) {
        const int rb = r0 + 8 * kh;
#pragma unroll
        for (int g = 0; g < 8; ++g)
          lse_buf[rb + g] = __logf(s[g]);
      }
    }
    __syncthreads();
    {
      const int c  = t & (NDIM - 1);
      const int r0 = (t >> 8) * 64;
#pragma unroll 4
      for (int r = r0; r < r0 + 64; ++r)
        tile[r * PITCH + c] -= lse_buf[r];   // lse_buf[r]: LDS broadcast
    }
    __syncthreads();

    // ================= column normalization (logsumexp over rows) ========
    // Waves 0..15: wave w owns cols [16w, 16w+16). B-operand = exp(col data),
    // A = ones -> D[m][n] = colsum_n. Pitch-257 makes these conflict-free.
    if (wv < 16) {
      const int c0 = wv * 16;
      const int n  = lane & 15;
      const int kr = lane >> 4;
      const float* colp = &tile[kr * PITCH + c0 + n];
      v8f acc0 = {}; v8f acc1 = {};
#pragma unroll 4
      for (int k = 0; k < NDIM; k += 8) {
        v2f b0, b1;
        b0.x = __expf(colp[(k + 0) * PITCH]);
        b0.y = __expf(colp[(k + 2) * PITCH]);
        b1.x = __expf(colp[(k + 4) * PITCH]);
        b1.y = __expf(colp[(k + 6) * PITCH]);
        acc0 = wmma_f32_16x16x4(vones, b0, acc0);
        acc1 = wmma_f32_16x16x4(vones, b1, acc1);
      }
      v8f s = acc0 + acc1;
      // Every vgpr holds colsum for column c0 + lane%16.
      if (lane < 16)
        lse_buf[c0 + n] = __logf(s[0]);
    }
    __syncthreads();
    {
      const int c  = t & (NDIM - 1);
      const int r0 = (t >> 8) * 64;
      const float cl = lse_buf[c];
#pragma unroll 4
      for (int r = r0; r < r0 + 64; ++r)
        tile[r * PITCH + c] -= cl;
    }
    __syncthreads();
  }

  // ---- out = exp(log_alpha); coalesced 128B stores per wave
  {
    float* outp = out + (size_t)b * (NDIM * NDIM);
    const int c  = t & (NDIM - 1);
    const int r0 = (t >> 8) * 64;
#pragma unroll 4
    for (int r = r0; r < r0 + 64; ++r)
      outp[r * NDIM + c] = __expf(tile[r * PITCH + c]);
  }
}

extern "C" void kernel_launch(void* const* d_in, const int* in_sizes, int n_in,
                              void* d_out, int out_size, void* d_ws,
                              size_t ws_size, hipStream_t stream) {
  const float* alpha = (const float*)d_in[0];
  float* out = (float*)d_out;
  const int B = in_sizes[0] / (NDIM * NDIM);  // = 16
  GumbelSinkhorn_gfx1250_kernel<<<dim3(B), dim3(1024), 0, stream>>>(alpha, out);
  (void)n_in; (void)out_size; (void)d_ws; (void)ws_size;
}